// WGAAE_Encoder_12463995093545
// MI455X (gfx1250) — compile-verified
//
#include <hip/hip_runtime.h>
#include <cmath>

#define HEADS 4
#define RMIN_F 2.2e-10f

typedef __attribute__((ext_vector_type(2))) float v2f;
typedef __attribute__((ext_vector_type(8))) float v8f;

__device__ __forceinline__ float softplus_f(float x) {
  return (x > 20.0f) ? x : log1pf(expf(x));
}
// monotone float<->uint encoding for atomic max on floats (incl. negatives)
__device__ __forceinline__ unsigned enc_f(float x) {
  unsigned b = __float_as_uint(x);
  return (b & 0x80000000u) ? ~b : (b | 0x80000000u);
}
__device__ __forceinline__ float dec_f(unsigned e) {
  return (e & 0x80000000u) ? __uint_as_float(e & 0x7FFFFFFFu) : __uint_as_float(~e);
}

// 32-bit LDS offset of a __shared__ object (generic -> addrspace(3) -> int)
__device__ __forceinline__ unsigned lds_off(const void* p) {
  return (unsigned)(uintptr_t)(__attribute__((address_space(3))) const void*)p;
}
// CDNA5 async copy global -> LDS (B32), tracked by ASYNCcnt, per-lane dsaddr
__device__ __forceinline__ void async_g2lds_b32(unsigned lds_addr,
                                                const float* gaddr) {
  asm volatile("global_load_async_to_lds_b32 %0, %1, off"
               :
               : "v"(lds_addr), "v"(gaddr)
               : "memory");
}

// ---------------------------------------------------------------------------
// Generic LDS-tiled fp32 WMMA GEMM: out = epilogue(A[MxK] * B[KxN] (+bias))
// Block tile 64x32, 8 waves (4x2 grid of 16x16 WMMA tiles). K stepped by 32
// (8 x v_wmma_f32_16x16x4_f32 per step). Tiles staged with double-buffered
// GLOBAL_LOAD_ASYNC_TO_LDS_B32 (ASYNCcnt) to overlap HBM/L2 latency with WMMA.
// mode 0: out = acc + bias                        (row-major)
// mode 1: out = BN(Cin + acc + bias)              (skip-add + BatchNorm eval)
// mode 2: out = softplus(acc + bias)
// mode 3: out^T = clip(softplus(acc+bias),0.1,1e3)    (store [N x M])
// mode 4: out^T = max(softplus(acc+bias), RMIN)       (store [N x M])
// ---------------------------------------------------------------------------
#define BM 64
#define BN 32
#define KT 32

__global__ __launch_bounds__(256) void gemm_wmma(
    const float* __restrict__ A, const float* __restrict__ B,
    const float* __restrict__ bias, const float* __restrict__ Cin,
    float* __restrict__ out, int M, int K, int Ncols, int mode,
    const float* __restrict__ bn_g, const float* __restrict__ bn_b,
    const float* __restrict__ bn_mu, const float* __restrict__ bn_va) {
  __shared__ float As[2][BM][KT + 1];
  __shared__ float Bs[2][KT][BN + 1];
  const int tid = threadIdx.x;
  const int mBase = blockIdx.x * BM;
  const int nBase = blockIdx.y * BN;
  const int wave = tid >> 5, lane = tid & 31;
  const int wm = wave >> 1, wn = wave & 1;  // 4 x 2 wave grid
  const int row16 = lane & 15, khalf = lane >> 4;
  v8f acc = {};

  // issue the 12 per-wave async tile loads (8 for A 64x32, 4 for B 32x32)
  auto issue_tile = [&](int k0, int buf) {
#pragma unroll
    for (int i = 0; i < 8; ++i) {
      int idx = tid + i * 256;
      int r = idx >> 5, c = idx & 31;
      int gr = mBase + r;
      if (gr >= M) gr = M - 1;  // clamp: tail rows never stored, no branch
      async_g2lds_b32(lds_off(&As[buf][r][c]), &A[(size_t)gr * K + (k0 + c)]);
    }
#pragma unroll
    for (int i = 0; i < 4; ++i) {
      int idx = tid + i * 256;
      int r = idx >> 5, c = idx & 31;
      async_g2lds_b32(lds_off(&Bs[buf][r][c]),
                      &B[(size_t)(k0 + r) * Ncols + (nBase + c)]);
    }
  };

  issue_tile(0, 0);
  const int nsteps = K / KT;
  for (int s = 0; s < nsteps; ++s) {
    const int buf = s & 1;
    if (s + 1 < nsteps) {
      issue_tile((s + 1) * KT, buf ^ 1);
      // own tile-s loads (issued before the 12 above) are done at cnt<=12
      asm volatile("s_wait_asynccnt 12" ::: "memory");
    } else {
      asm volatile("s_wait_asynccnt 0" ::: "memory");
    }
    __syncthreads();  // tile visible workgroup-wide
#pragma unroll
    for (int kk = 0; kk < KT; kk += 4) {
      // A 16x4 frag: lanes 0-15 K={0,1}, lanes 16-31 K={2,3}; M = lane%16
      // B 4x16 frag: mirrored; N = lane%16
      v2f a, b;
      a.x = As[buf][wm * 16 + row16][kk + khalf * 2 + 0];
      a.y = As[buf][wm * 16 + row16][kk + khalf * 2 + 1];
      b.x = Bs[buf][kk + khalf * 2 + 0][wn * 16 + row16];
      b.y = Bs[buf][kk + khalf * 2 + 1][wn * 16 + row16];
      acc = __builtin_amdgcn_wmma_f32_16x16x4_f32(false, a, false, b, (short)0,
                                                  acc, false, false);
    }
    __syncthreads();  // protect buf before it is refilled next-next step
  }

  const int col = nBase + wn * 16 + row16;
  const float bv = bias ? bias[col] : 0.0f;
#pragma unroll
  for (int r = 0; r < 8; ++r) {
    // C/D layout: VGPR r -> M = khalf*8 + r, N = lane%16
    int m = mBase + wm * 16 + khalf * 8 + r;
    if (m >= M) continue;
    float v = acc[r] + bv;
    if (mode == 0) {
      out[(size_t)m * Ncols + col] = v;
    } else if (mode == 1) {
      float gsum = Cin[(size_t)m * Ncols + col] + v;
      out[(size_t)m * Ncols + col] =
          (gsum - bn_mu[col]) * rsqrtf(bn_va[col] + 1e-5f) * bn_g[col] +
          bn_b[col];
    } else if (mode == 2) {
      out[(size_t)m * Ncols + col] = softplus_f(v);
    } else if (mode == 3) {
      float kv = softplus_f(v);
      out[(size_t)col * M + m] = fminf(fmaxf(kv, 0.1f), 1000.0f);
    } else {
      float lv = softplus_f(v);
      out[(size_t)col * M + m] = fmaxf(lv, RMIN_F);
    }
  }
}

// per-node attention logits: asrc[n,h] = sum_d h[n,h,d]*att_s[h,d] (same adst)
__global__ void alpha_kernel(const float* __restrict__ h,
                             const float* __restrict__ att_s,
                             const float* __restrict__ att_d,
                             float* __restrict__ asrc, float* __restrict__ adst,
                             int Nn, int D) {
  int i = blockIdx.x * blockDim.x + threadIdx.x;
  if (i >= Nn * HEADS) return;
  int n = i >> 2, hh = i & 3;
  const float* hp = h + (size_t)n * (HEADS * D) + hh * D;
  float s = 0.f, d = 0.f;
  for (int j = 0; j < D; ++j) {
    float v = hp[j];
    s += v * att_s[hh * D + j];
    d += v * att_d[hh * D + j];
  }
  asrc[i] = s;
  adst[i] = d;
}

__global__ void fill_u32(unsigned* __restrict__ p, unsigned v, size_t n) {
  size_t i = (size_t)blockIdx.x * blockDim.x + threadIdx.x;
  if (i < n) p[i] = v;
}

__device__ __forceinline__ void edge_sd(const int* ei, int E_, int e, int& s,
                                        int& d) {
  if (e < E_) {
    s = ei[e];
    d = ei[E_ + e];
  } else {
    s = e - E_;  // self loop
    d = e - E_;
  }
}

// pass A: segment max of leaky_relu(asrc[src]+adst[dst]) over dst
__global__ void edge_max_k(const int* __restrict__ ei, int E_, int ET,
                           const float* __restrict__ asrc,
                           const float* __restrict__ adst,
                           unsigned* __restrict__ menc) {
  int e = blockIdx.x * blockDim.x + threadIdx.x;
  if (e >= ET) return;
  int s, d;
  edge_sd(ei, E_, e, s, d);
#pragma unroll
  for (int hh = 0; hh < HEADS; ++hh) {
    float v = asrc[s * HEADS + hh] + adst[d * HEADS + hh];
    v = (v > 0.f) ? v : 0.2f * v;
    atomicMax(&menc[d * HEADS + hh], enc_f(v));
  }
}

// pass B: p = exp(e - m[dst]); z[dst] += p
__global__ void edge_exp_k(const int* __restrict__ ei, int E_, int ET,
                           const float* __restrict__ asrc,
                           const float* __restrict__ adst,
                           const unsigned* __restrict__ menc,
                           float* __restrict__ p, float* __restrict__ z) {
  int e = blockIdx.x * blockDim.x + threadIdx.x;
  if (e >= ET) return;
  int s, d;
  edge_sd(ei, E_, e, s, d);
#pragma unroll
  for (int hh = 0; hh < HEADS; ++hh) {
    float v = asrc[s * HEADS + hh] + adst[d * HEADS + hh];
    v = (v > 0.f) ? v : 0.2f * v;
    float pe = expf(v - dec_f(menc[d * HEADS + hh]));
    p[(size_t)e * HEADS + hh] = pe;
    atomicAdd(&z[d * HEADS + hh], pe);
  }
}

// pass C: agg[dst] += h[src] * alpha   (one wave32 per edge; L2-resident adds)
__global__ void edge_agg_k(const int* __restrict__ ei, int E_, int ET,
                           const float* __restrict__ hsrc,
                           const float* __restrict__ p,
                           const float* __restrict__ z,
                           float* __restrict__ agg, int HC) {
  int gw = (int)(((size_t)blockIdx.x * blockDim.x + threadIdx.x) >> 5);
  int lane = threadIdx.x & 31;
  if (gw >= ET) return;
  int s, d;
  edge_sd(ei, E_, gw, s, d);
  int D = HC >> 2;
  float al[HEADS];
#pragma unroll
  for (int hh = 0; hh < HEADS; ++hh)
    al[hh] = p[(size_t)gw * HEADS + hh] / (z[d * HEADS + hh] + 1e-16f);
  __builtin_prefetch(&hsrc[(size_t)s * HC], 0, 0);  // global_prefetch_b8
  for (int c = lane; c < HC; c += 32) {
    int hh = c / D;
    atomicAdd(&agg[(size_t)d * HC + c], hsrc[(size_t)s * HC + c] * al[hh]);
  }
}

__global__ void add_bias_k(float* __restrict__ g, const float* __restrict__ b,
                           size_t total, int Cmask) {
  size_t i = (size_t)blockIdx.x * blockDim.x + threadIdx.x;
  if (i < total) g[i] += b[i & Cmask];  // C is a power of two
}

// Weibull head: l /= exp(lgamma(1+1/k)); theta = clip(mean_s l*(-log(1-u))^(1/k))
__global__ void theta_kernel(const float* __restrict__ eps,
                             const float* __restrict__ k_out,
                             float* __restrict__ l_out,
                             float* __restrict__ th_out, int Kd, int Nn) {
  size_t i = (size_t)blockIdx.x * blockDim.x + threadIdx.x;
  size_t tot = (size_t)Kd * Nn;
  if (i >= tot) return;
  float k = k_out[i];
  float invk = 1.0f / k;
  float l = l_out[i] * expf(-lgammaf(1.0f + invk));
  l_out[i] = l;
  float acc = 0.f;
#pragma unroll
  for (int s = 0; s < 10; ++s) {
    float u = eps[(size_t)s * tot + i];
    float t = -logf(fmaxf(1.0f - u, RMIN_F));
    acc += powf(t, invk);
  }
  float th = l * acc * 0.1f;
  th_out[i] = fminf(fmaxf(th, RMIN_F), 1000.0f);
}

// ---------------------------------------------------------------------------
extern "C" void kernel_launch(void* const* d_in, const int* in_sizes, int n_in,
                              void* d_out, int out_size, void* d_ws,
                              size_t ws_size, hipStream_t stream) {
  const float* x = (const float*)d_in[0];
  const int* ei = (const int*)d_in[1];
  const float* eps0 = (const float*)d_in[2];
  const float* eps1 = (const float*)d_in[3];
  const float* W_gat0 = (const float*)d_in[4];
  const float* att_s0 = (const float*)d_in[5];
  const float* att_d0 = (const float*)d_in[6];
  const float* b_gat0 = (const float*)d_in[7];
  const float* W_skip0 = (const float*)d_in[8];
  const float* b_skip0 = (const float*)d_in[9];
  const float* gamma0 = (const float*)d_in[10];
  const float* beta0 = (const float*)d_in[11];
  const float* mean0 = (const float*)d_in[12];
  const float* var0 = (const float*)d_in[13];
  const float* W_mlp0 = (const float*)d_in[14];
  const float* b_mlp0 = (const float*)d_in[15];
  const float* W_shape0 = (const float*)d_in[16];
  const float* b_shape0 = (const float*)d_in[17];
  const float* W_scale0 = (const float*)d_in[18];
  const float* b_scale0 = (const float*)d_in[19];
  const float* W_gat1 = (const float*)d_in[20];
  const float* att_s1 = (const float*)d_in[21];
  const float* att_d1 = (const float*)d_in[22];
  const float* b_gat1 = (const float*)d_in[23];
  const float* W_skip1 = (const float*)d_in[24];
  const float* b_skip1 = (const float*)d_in[25];
  const float* gamma1 = (const float*)d_in[26];
  const float* beta1 = (const float*)d_in[27];
  const float* mean1 = (const float*)d_in[28];
  const float* var1 = (const float*)d_in[29];
  const float* W_mlp1 = (const float*)d_in[30];
  const float* b_mlp1 = (const float*)d_in[31];
  const float* W_shape1 = (const float*)d_in[32];
  const float* b_shape1 = (const float*)d_in[33];
  const float* W_scale1 = (const float*)d_in[34];
  const float* b_scale1 = (const float*)d_in[35];

  const int Nn = in_sizes[0] / 256;  // 50000
  const int E_ = in_sizes[1] / 2;    // 400000
  const int ET = E_ + Nn;            // with self loops

  // workspace layout (floats), with aliasing of dead buffers
  float* ws = (float*)d_ws;
  size_t o = 0;
  float* hgat0 = ws + o; o += (size_t)Nn * 256;
  float* agg0  = ws + o; o += (size_t)Nn * 256;
  float* h0    = ws + o; o += (size_t)Nn * 256;
  float* h1    = ws + o; o += (size_t)Nn * 128;
  float* asrc  = ws + o; o += (size_t)Nn * 4;
  float* adst  = ws + o; o += (size_t)Nn * 4;
  unsigned* menc = (unsigned*)(ws + o); o += (size_t)Nn * 4;
  float* zbuf  = ws + o; o += (size_t)Nn * 4;
  float* pbuf  = ws + o; o += (size_t)ET * 4;
  float* agg1  = agg0;                         // agg0 dead after h0 GEMM
  float* hgat1 = agg0 + (size_t)Nn * 128;
  float* f0    = hgat0;                        // hgat0 dead after L0 aggregate
  float* f1    = hgat0 + (size_t)Nn * 128;

  float* outp = (float*)d_out;
  float* th0 = outp;
  float* th1 = outp + (size_t)128 * Nn;
  float* k0  = outp + (size_t)192 * Nn;
  float* k1  = outp + (size_t)320 * Nn;
  float* l0  = outp + (size_t)384 * Nn;
  float* l1  = outp + (size_t)512 * Nn;

  dim3 blk(256);
  auto g1d = [](size_t n) { return dim3((unsigned)((n + 255) / 256)); };
  const unsigned mgrid = (unsigned)((Nn + BM - 1) / BM);
  const unsigned ENC_NEG_INF = 0x007FFFFFu;  // enc_f(-inf)

  // ---------------- layer 0 GAT ----------------
  gemm_wmma<<<dim3(mgrid, 256 / BN), blk, 0, stream>>>(
      x, W_gat0, nullptr, nullptr, hgat0, Nn, 256, 256, 0, nullptr, nullptr,
      nullptr, nullptr);
  alpha_kernel<<<g1d((size_t)Nn * 4), blk, 0, stream>>>(hgat0, att_s0, att_d0,
                                                        asrc, adst, Nn, 64);
  fill_u32<<<g1d((size_t)Nn * 4), blk, 0, stream>>>(menc, ENC_NEG_INF,
                                                    (size_t)Nn * 4);
  fill_u32<<<g1d((size_t)Nn * 4), blk, 0, stream>>>((unsigned*)zbuf, 0u,
                                                    (size_t)Nn * 4);
  fill_u32<<<g1d((size_t)Nn * 256), blk, 0, stream>>>((unsigned*)agg0, 0u,
                                                      (size_t)Nn * 256);
  edge_max_k<<<g1d(ET), blk, 0, stream>>>(ei, E_, ET, asrc, adst, menc);
  edge_exp_k<<<g1d(ET), blk, 0, stream>>>(ei, E_, ET, asrc, adst, menc, pbuf,
                                          zbuf);
  edge_agg_k<<<g1d((size_t)ET * 32), blk, 0, stream>>>(ei, E_, ET, hgat0, pbuf,
                                                       zbuf, agg0, 256);
  add_bias_k<<<g1d((size_t)Nn * 256), blk, 0, stream>>>(agg0, b_gat0,
                                                        (size_t)Nn * 256, 255);
  gemm_wmma<<<dim3(mgrid, 256 / BN), blk, 0, stream>>>(
      agg0, W_skip0, b_skip0, agg0, h0, Nn, 256, 256, 1, gamma0, beta0, mean0,
      var0);

  // ---------------- layer 1 GAT ----------------
  gemm_wmma<<<dim3(mgrid, 128 / BN), blk, 0, stream>>>(
      h0, W_gat1, nullptr, nullptr, hgat1, Nn, 256, 128, 0, nullptr, nullptr,
      nullptr, nullptr);
  alpha_kernel<<<g1d((size_t)Nn * 4), blk, 0, stream>>>(hgat1, att_s1, att_d1,
                                                        asrc, adst, Nn, 32);
  fill_u32<<<g1d((size_t)Nn * 4), blk, 0, stream>>>(menc, ENC_NEG_INF,
                                                    (size_t)Nn * 4);
  fill_u32<<<g1d((size_t)Nn * 4), blk, 0, stream>>>((unsigned*)zbuf, 0u,
                                                    (size_t)Nn * 4);
  fill_u32<<<g1d((size_t)Nn * 128), blk, 0, stream>>>((unsigned*)agg1, 0u,
                                                      (size_t)Nn * 128);
  edge_max_k<<<g1d(ET), blk, 0, stream>>>(ei, E_, ET, asrc, adst, menc);
  edge_exp_k<<<g1d(ET), blk, 0, stream>>>(ei, E_, ET, asrc, adst, menc, pbuf,
                                          zbuf);
  edge_agg_k<<<g1d((size_t)ET * 32), blk, 0, stream>>>(ei, E_, ET, hgat1, pbuf,
                                                       zbuf, agg1, 128);
  add_bias_k<<<g1d((size_t)Nn * 128), blk, 0, stream>>>(agg1, b_gat1,
                                                        (size_t)Nn * 128, 127);
  gemm_wmma<<<dim3(mgrid, 128 / BN), blk, 0, stream>>>(
      agg1, W_skip1, b_skip1, agg1, h1, Nn, 128, 128, 1, gamma1, beta1, mean1,
      var1);

  // ---------------- MLPs + Weibull heads ----------------
  gemm_wmma<<<dim3(mgrid, 128 / BN), blk, 0, stream>>>(
      h0, W_mlp0, b_mlp0, nullptr, f0, Nn, 256, 128, 2, nullptr, nullptr,
      nullptr, nullptr);
  gemm_wmma<<<dim3(mgrid, 64 / BN), blk, 0, stream>>>(
      h1, W_mlp1, b_mlp1, nullptr, f1, Nn, 128, 64, 2, nullptr, nullptr,
      nullptr, nullptr);

  gemm_wmma<<<dim3(mgrid, 128 / BN), blk, 0, stream>>>(
      f0, W_shape0, b_shape0, nullptr, k0, Nn, 128, 128, 3, nullptr, nullptr,
      nullptr, nullptr);
  gemm_wmma<<<dim3(mgrid, 128 / BN), blk, 0, stream>>>(
      f0, W_scale0, b_scale0, nullptr, l0, Nn, 128, 128, 4, nullptr, nullptr,
      nullptr, nullptr);
  theta_kernel<<<g1d((size_t)128 * Nn), blk, 0, stream>>>(eps0, k0, l0, th0,
                                                          128, Nn);

  gemm_wmma<<<dim3(mgrid, 64 / BN), blk, 0, stream>>>(
      f1, W_shape1, b_shape1, nullptr, k1, Nn, 64, 64, 3, nullptr, nullptr,
      nullptr, nullptr);
  gemm_wmma<<<dim3(mgrid, 64 / BN), blk, 0, stream>>>(
      f1, W_scale1, b_scale1, nullptr, l1, Nn, 64, 64, 4, nullptr, nullptr,
      nullptr, nullptr);
  theta_kernel<<<g1d((size_t)64 * Nn), blk, 0, stream>>>(eps1, k1, l1, th1, 64,
                                                         Nn);
}